// _HomogeneousRGCNWrapper_60352880443451
// MI455X (gfx1250) — compile-verified
//
#include <hip/hip_runtime.h>
#include <hip/hip_bf16.h>

#define N_NODES   50000
#define N_EDGES   800000
#define D_FEAT    128
#define N_REL     4
#define N_GRAPHS  128
#define N_CLASSES 16
#define K_TOTAL   (N_REL * D_FEAT + D_FEAT)   // 640
#define M_TILES   5                           // M tiles per wave (80 rows/block)
#define M_BLOCK   (M_TILES * 16)              // 80; 50000 = 625 * 80

typedef __attribute__((ext_vector_type(2))) float v2f;
typedef __attribute__((ext_vector_type(8))) float v8f;

// ---------------------------------------------------------------------------
// Zero a float buffer (count must be a multiple of 4). Grid-stride float4.
// ---------------------------------------------------------------------------
__global__ void k_zero_f32(float* __restrict__ p, long long count4) {
    long long i = (long long)blockIdx.x * blockDim.x + threadIdx.x;
    long long stride = (long long)gridDim.x * blockDim.x;
    float4 z = make_float4(0.f, 0.f, 0.f, 0.f);
    for (; i < count4; i += stride)
        ((float4*)p)[i] = z;
}

// ---------------------------------------------------------------------------
// Count edges per (dst, relation) segment. cnt is [N_NODES * N_REL] f32.
// ---------------------------------------------------------------------------
__global__ void k_edge_count(const int* __restrict__ dst,
                             const int* __restrict__ etype,
                             float* __restrict__ cnt) {
    int e = blockIdx.x * blockDim.x + threadIdx.x;
    if (e >= N_EDGES) return;
    atomicAdd(&cnt[dst[e] * N_REL + etype[e]], 1.0f);
}

// ---------------------------------------------------------------------------
// Pack combined B = [W_rel(512x128); W_root(128x128)] into K-pair-interleaved
// layout: Wpk[(k/2)*128 + col] = { B[k][col], B[k+1][col] }  (k even).
// Matches the f32-WMMA per-lane B operand -> one b64 load per K-step.
// ---------------------------------------------------------------------------
__global__ void k_pack_w(const float* __restrict__ Wrel,   // [512,128]
                         const float* __restrict__ Wroot,  // [128,128]
                         float2* __restrict__ Wpk) {       // [320*128]
    int t = blockIdx.x * blockDim.x + threadIdx.x;
    if (t >= (K_TOTAL / 2) * D_FEAT) return;
    int khalf = t >> 7;            // 0..319
    int col = t & (D_FEAT - 1);
    int k = khalf * 2;             // even; k and k+1 in same region
    float bx, by;
    if (k < N_REL * D_FEAT) {
        bx = Wrel[(long long)k * D_FEAT + col];
        by = Wrel[(long long)(k + 1) * D_FEAT + col];
    } else {
        int kr = k - N_REL * D_FEAT;
        bx = Wroot[(long long)kr * D_FEAT + col];
        by = Wroot[(long long)(kr + 1) * D_FEAT + col];
    }
    Wpk[t] = make_float2(bx, by);
}

// ---------------------------------------------------------------------------
// Scatter-add mean-normalized source features into A[N][640] cols 0..511
// (relation blocks). One wave per edge; each lane handles 4 dims (float4).
// A (128 MB) is L2-resident on MI455X (192 MB L2), so atomics stay on-chip.
// ---------------------------------------------------------------------------
__global__ void k_scatter(const float* __restrict__ hin,    // [N, 128]
                          const int* __restrict__ src,
                          const int* __restrict__ dst,
                          const int* __restrict__ etype,
                          const float* __restrict__ cnt,    // [N*R]
                          float* __restrict__ A) {          // [N, 640]
    int e = blockIdx.x * (blockDim.x >> 5) + (threadIdx.x >> 5);
    if (e >= N_EDGES) return;
    int lane = threadIdx.x & 31;
    int s = src[e];
    int d = dst[e];
    int r = etype[e];
    float c = cnt[d * N_REL + r];
    float norm = 1.0f / fmaxf(c, 1.0f);
    const float4 v = *(const float4*)(hin + (long long)s * D_FEAT + lane * 4);
    float* out = A + (long long)d * K_TOTAL + r * D_FEAT + lane * 4;
    atomicAdd(out + 0, v.x * norm);
    atomicAdd(out + 1, v.y * norm);
    atomicAdd(out + 2, v.z * norm);
    atomicAdd(out + 3, v.w * norm);
}

// ---------------------------------------------------------------------------
// Copy h_in into A cols 512..639 (the W_root self-term of the K dimension).
// ---------------------------------------------------------------------------
__global__ void k_copy_tail(const float* __restrict__ hin,  // [N, 128]
                            float* __restrict__ A) {        // [N, 640]
    long long i = (long long)blockIdx.x * blockDim.x + threadIdx.x;
    if (i >= (long long)N_NODES * D_FEAT) return;
    int row = (int)(i >> 7);
    int d = (int)(i & (D_FEAT - 1));
    A[(long long)row * K_TOTAL + N_REL * D_FEAT + d] = hin[i];
}

// ---------------------------------------------------------------------------
// Branch-free RGCN layer GEMM via V_WMMA_F32_16X16X4_F32:
//   hout[n, f] = relu?( A[n, 0:640] . B[0:640, f] + bias[f] )
// Block = 256 threads = 8 waves; each wave owns 5 M-tiles x one 16-col tile.
// Block covers 80 rows x all 128 cols. Grid = 50000/80 = 625.
//
// f32 WMMA lane layout (ISA 7.12.2):
//   A 16x4 : lane(0..15) row M=lane, V0=K0 V1=K1 ; lane(16..31) K2,K3
//   B 4x16 : lane(0..15) col N=lane, V0=K0 V1=K1 ; lane(16..31) K2,K3
//   C/D    : VGPR i -> M = i + 8*(lane>>4), N = lane&15
// ---------------------------------------------------------------------------
__global__ void k_rgcn_gemm(const float* __restrict__ A,     // [N, 640]
                            const float2* __restrict__ Wpk,  // [320, 128]
                            const float* __restrict__ bias,  // [128]
                            float* __restrict__ hout,        // [N, 128]
                            int relu) {
    const int wave  = threadIdx.x >> 5;   // 0..7 -> N tile
    const int lane  = threadIdx.x & 31;
    const int lmod  = lane & 15;
    const int lhalf = lane >> 4;          // 0 or 1

    const int m0 = blockIdx.x * M_BLOCK;  // row base of this slab
    const int n0 = wave * 16;             // col base of this wave's tiles
    const int bcol = n0 + lmod;           // B col owned by this lane

    const float* aRow[M_TILES];
#pragma unroll
    for (int t = 0; t < M_TILES; ++t)
        aRow[t] = A + (long long)(m0 + t * 16 + lmod) * K_TOTAL;

    v8f acc[M_TILES];
#pragma unroll
    for (int t = 0; t < M_TILES; ++t)
        acc[t] = (v8f){};

#pragma unroll 4
    for (int kk = 0; kk < K_TOTAL; kk += 4) {
        const int k = kk + 2 * lhalf;     // this lane's K pair: {k, k+1}
        const float2 bb = Wpk[(k >> 1) * D_FEAT + bcol];
        v2f b; b.x = bb.x; b.y = bb.y;
#pragma unroll
        for (int t = 0; t < M_TILES; ++t) {
            const float2 aa = *(const float2*)(aRow[t] + k);
            v2f a; a.x = aa.x; a.y = aa.y;
            acc[t] = __builtin_amdgcn_wmma_f32_16x16x4_f32(
                false, a, false, b, (short)0, acc[t], false, false);
        }
    }

    // Writeback per C/D layout.
    const float bv = bias[bcol];
#pragma unroll
    for (int t = 0; t < M_TILES; ++t) {
        const int outRow0 = m0 + t * 16 + lhalf * 8;
#pragma unroll
        for (int i = 0; i < 8; ++i) {
            float v = acc[t][i] + bv;
            if (relu) v = fmaxf(v, 0.0f);
            hout[(long long)(outRow0 + i) * D_FEAT + bcol] = v;
        }
    }
}

// ---------------------------------------------------------------------------
// Global mean pool: node counts per graph.
// ---------------------------------------------------------------------------
__global__ void k_pool_count(const int* __restrict__ batch,
                             float* __restrict__ gcnt) {
    int n = blockIdx.x * blockDim.x + threadIdx.x;
    if (n >= N_NODES) return;
    atomicAdd(&gcnt[batch[n]], 1.0f);
}

// Global mean pool: per-dim sums.
__global__ void k_pool_sum(const float* __restrict__ h,     // [N, 128]
                           const int* __restrict__ batch,
                           float* __restrict__ gsum) {      // [G, 128]
    long long i = (long long)blockIdx.x * blockDim.x + threadIdx.x;
    if (i >= (long long)N_NODES * D_FEAT) return;
    int node = (int)(i >> 7);
    int d = (int)(i & (D_FEAT - 1));
    atomicAdd(&gsum[batch[node] * D_FEAT + d], h[i]);
}

// ---------------------------------------------------------------------------
// Classifier: out[g, c] = mean(h)[g] . Wcls[:, c] + bcls[c]
// ---------------------------------------------------------------------------
__global__ void k_classify(const float* __restrict__ gsum,  // [G, 128]
                           const float* __restrict__ gcnt,  // [G]
                           const float* __restrict__ Wcls,  // [128, 16]
                           const float* __restrict__ bcls,  // [16]
                           float* __restrict__ out) {       // [G, 16]
    int t = blockIdx.x * blockDim.x + threadIdx.x;
    if (t >= N_GRAPHS * N_CLASSES) return;
    int g = t >> 4;
    int c = t & (N_CLASSES - 1);
    float inv = 1.0f / fmaxf(gcnt[g], 1.0f);
    float s = 0.0f;
#pragma unroll 8
    for (int d = 0; d < D_FEAT; ++d)
        s += gsum[g * D_FEAT + d] * Wcls[d * N_CLASSES + c];
    out[t] = s * inv + bcls[c];
}

// ---------------------------------------------------------------------------
// Host-side orchestration (graph-capture safe: only kernel launches on stream)
// ---------------------------------------------------------------------------
extern "C" void kernel_launch(void* const* d_in, const int* in_sizes, int n_in,
                              void* d_out, int out_size, void* d_ws, size_t ws_size,
                              hipStream_t stream) {
    const float* x       = (const float*)d_in[0];
    const int*   eidx    = (const int*)d_in[1];     // [2, E]
    const int*   src     = eidx;
    const int*   dst     = eidx + N_EDGES;
    const int*   etype   = (const int*)d_in[2];
    const int*   batch   = (const int*)d_in[3];
    const float* Wrel1   = (const float*)d_in[4];   // [4,128,128] -> [512,128]
    const float* Wroot1  = (const float*)d_in[5];
    const float* b1      = (const float*)d_in[6];
    const float* Wrel2   = (const float*)d_in[7];
    const float* Wroot2  = (const float*)d_in[8];
    const float* b2      = (const float*)d_in[9];
    const float* Wcls    = (const float*)d_in[10];
    const float* bcls    = (const float*)d_in[11];
    float* out = (float*)d_out;

    // Workspace partition (256-byte aligned slices).
    char* ws = (char*)d_ws;
    size_t off = 0;
    auto alloc = [&](size_t bytes) {
        char* p = ws + off;
        off += (bytes + 255) & ~(size_t)255;
        return p;
    };
    float*  Abuf = (float*)alloc((size_t)N_NODES * K_TOTAL * 4);    // 128 MB
    float*  cnt  = (float*)alloc((size_t)N_NODES * N_REL * 4);      // 0.8 MB
    float*  h1   = (float*)alloc((size_t)N_NODES * D_FEAT * 4);     // 25.6 MB
    float*  h2   = (float*)alloc((size_t)N_NODES * D_FEAT * 4);     // 25.6 MB
    float2* Wpk1 = (float2*)alloc((size_t)(K_TOTAL / 2) * D_FEAT * 8);
    float2* Wpk2 = (float2*)alloc((size_t)(K_TOTAL / 2) * D_FEAT * 8);
    float*  gsum = (float*)alloc((size_t)N_GRAPHS * D_FEAT * 4);
    float*  gcnt = (float*)alloc((size_t)N_GRAPHS * 4);
    (void)ws_size; (void)n_in; (void)in_sizes; (void)out_size;

    const int ZB = 1024;  // blocks for zero kernels (grid-stride)
    const int PACK_N = (K_TOTAL / 2) * D_FEAT;           // 40960
    const long long ND = (long long)N_NODES * D_FEAT;    // 6.4M

    // --- Edge-degree counts + weight prepack (shared setup) ---
    k_zero_f32<<<ZB, 256, 0, stream>>>(cnt, (long long)(N_NODES * N_REL) / 4);
    k_edge_count<<<(N_EDGES + 255) / 256, 256, 0, stream>>>(dst, etype, cnt);
    k_pack_w<<<(PACK_N + 255) / 256, 256, 0, stream>>>(Wrel1, Wroot1, Wpk1);
    k_pack_w<<<(PACK_N + 255) / 256, 256, 0, stream>>>(Wrel2, Wroot2, Wpk2);

    // --- Layer 1: x -> h1 ---
    k_zero_f32<<<ZB, 256, 0, stream>>>(Abuf, (long long)N_NODES * K_TOTAL / 4);
    k_scatter<<<(N_EDGES + 7) / 8, 256, 0, stream>>>(x, src, dst, etype, cnt, Abuf);
    k_copy_tail<<<(int)((ND + 255) / 256), 256, 0, stream>>>(x, Abuf);
    k_rgcn_gemm<<<N_NODES / M_BLOCK, 256, 0, stream>>>(Abuf, Wpk1, b1, h1, 1);

    // --- Layer 2: h1 -> h2 ---
    k_zero_f32<<<ZB, 256, 0, stream>>>(Abuf, (long long)N_NODES * K_TOTAL / 4);
    k_scatter<<<(N_EDGES + 7) / 8, 256, 0, stream>>>(h1, src, dst, etype, cnt, Abuf);
    k_copy_tail<<<(int)((ND + 255) / 256), 256, 0, stream>>>(h1, Abuf);
    k_rgcn_gemm<<<N_NODES / M_BLOCK, 256, 0, stream>>>(Abuf, Wpk2, b2, h2, 1);

    // --- Global mean pool ---
    k_zero_f32<<<1, 256, 0, stream>>>(gsum, (long long)(N_GRAPHS * D_FEAT) / 4);
    k_zero_f32<<<1, 32, 0, stream>>>(gcnt, (long long)N_GRAPHS / 4);
    k_pool_count<<<(N_NODES + 255) / 256, 256, 0, stream>>>(batch, gcnt);
    k_pool_sum<<<(int)((ND + 255) / 256), 256, 0, stream>>>(h2, batch, gsum);

    // --- Classifier ---
    k_classify<<<(N_GRAPHS * N_CLASSES + 255) / 256, 256, 0, stream>>>(gsum, gcnt, Wcls, bcls, out);
}